// HypRegressionPoincare_75617194213415
// MI455X (gfx1250) — compile-verified
//
#include <hip/hip_runtime.h>
#include <math.h>

typedef __attribute__((ext_vector_type(2))) float v2f;
typedef __attribute__((ext_vector_type(8))) float v8f;

#define BSZ 8192
#define DSZ 128
#define PSZ 128

// ---------------- Kernel 0: per-p preprocessing ----------------
__device__ __forceinline__ float block_reduce_128(float v, float* sm) {
    int t = threadIdx.x;
    sm[t] = v;
    __syncthreads();
    for (int s = 64; s > 0; s >>= 1) {
        if (t < s) sm[t] += sm[t + s];
        __syncthreads();
    }
    float r = sm[0];
    __syncthreads();
    return r;
}

__global__ void hyp_prep_p(const float* __restrict__ weight,
                           const float* __restrict__ bias,
                           float* __restrict__ pneg, float* __restrict__ amat,
                           float* __restrict__ p2o, float* __restrict__ pao,
                           float* __restrict__ anormo, float* __restrict__ lampo) {
    __shared__ float sm[128];
    int p = blockIdx.x;
    int d = threadIdx.x;
    float bv = bias[p * DSZ + d];
    float n2 = block_reduce_128(bv * bv, sm);
    float nrm = fmaxf(sqrtf(n2), 1e-15f);
    const float maxn = 1.0f - 1e-5f;                 // (1 - PROJ_EPS)/sqrt(c), c=1
    float scale = (nrm > maxn) ? (maxn / nrm) : 1.0f;
    float bp = bv * scale;                            // projected bias
    float b2 = n2 * scale * scale;                    // ||bias_proj||^2
    float av = weight[p * DSZ + d] * (1.0f - b2);     // parallel-transported weight
    pneg[p * DSZ + d] = -bp;
    amat[p * DSZ + d] = av;
    float pav = block_reduce_128(-bp * av, sm);       // <p_neg, a>
    float a2  = block_reduce_128(av * av, sm);
    if (d == 0) {
        p2o[p]    = b2;                               // ||p_neg||^2 == b2
        pao[p]    = pav;
        anormo[p] = fmaxf(sqrtf(a2), 1e-15f);
        lampo[p]  = 2.0f / (1.0f - b2);
    }
}

// ---------------- Kernel 1: x2[b] = ||x_b||^2, one wave per row ----------------
__global__ void hyp_prep_x2(const float* __restrict__ x, float* __restrict__ x2) {
    int wid  = (blockIdx.x * blockDim.x + threadIdx.x) >> 5;
    int lane = threadIdx.x & 31;
    if (wid >= BSZ) return;
    const float* row = x + wid * DSZ;
    float s = 0.0f;
    #pragma unroll
    for (int i = 0; i < DSZ / 32; ++i) {
        float v = row[lane + 32 * i];
        s += v * v;
    }
    #pragma unroll
    for (int off = 16; off > 0; off >>= 1) s += __shfl_xor(s, off, 32);
    if (lane == 0) x2[wid] = s;
}

// ---------------- Kernel 2: dual WMMA GEMM + hyperbolic epilogue ----------------
__global__ __launch_bounds__(256) void hyp_main(
    const float* __restrict__ x, const float* __restrict__ pneg,
    const float* __restrict__ amat, const float* __restrict__ p2a,
    const float* __restrict__ paa, const float* __restrict__ anorma,
    const float* __restrict__ lampa, const float* __restrict__ x2a,
    float* __restrict__ out) {
    int lane = threadIdx.x & 31;
    int wave = threadIdx.x >> 5;
    int t     = blockIdx.x * 8 + wave;     // tile id
    int tileP = t & 7;                     // P / 16 = 8 column tiles
    int tileB = t >> 3;                    // 512 row tiles

    int row   = lane & 15;                 // A-row (lane<16) / B-col
    int khalf = (lane >> 4) * 2;           // K sub-offset per fp32 WMMA layout

    const float* xrow = x    + (tileB * 16 + row) * DSZ + khalf;
    const float* prow = pneg + (tileP * 16 + row) * DSZ + khalf;
    const float* arow = amat + (tileP * 16 + row) * DSZ + khalf;

    v8f accD = {};   // dot = x . p_neg^T
    v8f accX = {};   // xa  = x . a^T

    // Software-pipelined, fully unrolled K loop: fragments for step k+4 are
    // loaded while the WMMAs for step k execute.
    v2f Af0 = *(const v2f*)(xrow + 0);
    v2f Bp0 = *(const v2f*)(prow + 0);
    v2f Ba0 = *(const v2f*)(arow + 0);
    #pragma unroll
    for (int k = 0; k < DSZ; k += 4) {
        v2f Af1 = Af0, Bp1 = Bp0, Ba1 = Ba0;
        if (k + 4 < DSZ) {
            Af1 = *(const v2f*)(xrow + k + 4);
            Bp1 = *(const v2f*)(prow + k + 4);
            Ba1 = *(const v2f*)(arow + k + 4);
        }
        accD = __builtin_amdgcn_wmma_f32_16x16x4_f32(
            false, Af0, false, Bp0, (short)0, accD, false, false);
        accX = __builtin_amdgcn_wmma_f32_16x16x4_f32(
            false, Af0, false, Ba0, (short)0, accX, false, false);
        Af0 = Af1; Bp0 = Bp1; Ba0 = Ba1;
    }

    // Epilogue: hyperbolic signed-distance math (c = 1)
    int n = lane & 15;
    int p = tileP * 16 + n;
    float p2v = p2a[p];
    float pav = paa[p];
    float anv = anorma[p];
    float lpv = lampa[p];
    float cfx = 1.0f - p2v;
    int mbase = (lane >> 4) * 8;
    const float CLAMPV = 16.635532f;       // log(2/eps32)
    const float SMOOTH = 50.0f;

    #pragma unroll
    for (int i = 0; i < 8; ++i) {
        int b = tileB * 16 + mbase + i;
        float x2v = x2a[b];
        float dot = accD[i];
        float xav = accX[i];
        float cfp = 1.0f + 2.0f * dot + x2v;
        float den = 1.0f + 2.0f * dot + p2v * x2v;
        float id  = 1.0f / den;
        float suba = (cfp * pav + cfx * xav) * id;
        float sub2 = (cfp * cfp * p2v + 2.0f * cfp * cfx * dot + cfx * cfx * x2v) * id * id;
        float lsub = 2.0f / (1.0f - sub2);
        float arg  = lsub * suba / anv;
        // smooth_clamp(arg, -CLAMP, CLAMP, 50): stable softplus
        float z1  = SMOOTH * (arg + CLAMPV);
        float z2  = SMOOTH * (arg - CLAMPV);
        float sp1 = (fmaxf(z1, 0.0f) + log1pf(expf(-fabsf(z1)))) * (1.0f / SMOOTH);
        float sp2 = (fmaxf(z2, 0.0f) + log1pf(expf(-fabsf(z2)))) * (1.0f / SMOOTH);
        float cl  = -CLAMPV + sp1 - sp2;
        float dist = asinhf(cl);
        out[b * PSZ + p] = lpv * anv * dist;
    }
}

extern "C" void kernel_launch(void* const* d_in, const int* in_sizes, int n_in,
                              void* d_out, int out_size, void* d_ws, size_t ws_size,
                              hipStream_t stream) {
    const float* x      = (const float*)d_in[0];   // (8192, 128)
    const float* weight = (const float*)d_in[1];   // (128, 128)
    const float* bias   = (const float*)d_in[2];   // (128, 128)
    float* out = (float*)d_out;                    // (8192, 128)

    float* ws    = (float*)d_ws;
    float* pneg  = ws;               // 16384
    float* amat  = ws + 16384;       // 16384
    float* p2    = ws + 32768;       // 128
    float* pa    = ws + 32896;       // 128
    float* anorm = ws + 33024;       // 128
    float* lamp  = ws + 33152;       // 128
    float* x2    = ws + 33280;       // 8192   (total 41472 floats = 162 KB)

    hyp_prep_p<<<PSZ, 128, 0, stream>>>(weight, bias, pneg, amat, p2, pa, anorm, lamp);
    hyp_prep_x2<<<(BSZ * 32) / 256, 256, 0, stream>>>(x, x2);
    hyp_main<<<(BSZ / 16) * (PSZ / 16) / 8, 256, 0, stream>>>(
        x, pneg, amat, p2, pa, anorm, lamp, x2, out);
}